// Model_80693845557452
// MI455X (gfx1250) — compile-verified
//
#include <hip/hip_runtime.h>

typedef __attribute__((ext_vector_type(2))) float v2f;
typedef __attribute__((ext_vector_type(8))) float v8f;

#define BATCH 8
#define CIN   3
#define HIN   384
#define WIN   384
#define KK    9      // 3x3 kernel positions
#define CO1   18     // offset channels = KK*2
#define KRED  27     // CIN * KK reduction length
#define HO    382
#define WO    382
#define NPIX  (BATCH*HO*WO)
#define PIX_PER_BLOCK 256

// Fused deformable conv:
//  Stage 1: offsets = conv3x3(x, w1)+b1 via V_WMMA_F32_16X16X4_F32 (M=18 ch, K=27, N=pixels)
//  Stage 2: bilinear deformable sampling + 3x3x9 einsum with w2, +b2
__global__ __launch_bounds__(256) void deform_conv_fused(
    const float* __restrict__ x,  const float* __restrict__ w1,
    const float* __restrict__ b1, const float* __restrict__ w2,
    const float* __restrict__ b2, float* __restrict__ out)
{
    __shared__ float s_b1[CO1];
    __shared__ float s_w2[81];
    __shared__ float s_b2[3];
    __shared__ float s_off[PIX_PER_BLOCK * CO1];   // 18 offsets per pixel

    const int tid = threadIdx.x;
    if (tid < CO1)                 s_b1[tid]       = b1[tid];
    if (tid >= 32 && tid < 113)    s_w2[tid - 32]  = w2[tid - 32];
    if (tid >= 128 && tid < 131)   s_b2[tid - 128] = b2[tid - 128];
    __syncthreads();

    const int lane    = tid & 31;
    const int waveId  = tid >> 5;
    const int n       = lane & 15;          // N (pixel-in-tile) / A-matrix row
    const int hiHalf  = lane >> 4;          // lanes 16..31 carry K+2 for A and B
    const int blockPixBase = blockIdx.x * PIX_PER_BLOCK;
    const int wavePixBase  = blockPixBase + waveId * 32;

    // ---- Stage 1: preload A (w1) for all 7 K-chunks --------------------
    // A0: output channels 0..15 ; A1: channels 16..17 in rows 0..1, rest zero
    float a0[7][2], a1[7][2];
#pragma unroll
    for (int t = 0; t < 7; ++t) {
#pragma unroll
        for (int j = 0; j < 2; ++j) {
            int k  = 4 * t + 2 * hiHalf + j;
            int kc = (k < KRED) ? k : 0;
            float wa = w1[n * KRED + kc];
            float wb = (n < 2) ? w1[(16 + n) * KRED + kc] : 0.0f;
            a0[t][j] = (k < KRED) ? wa : 0.0f;
            a1[t][j] = (k < KRED) ? wb : 0.0f;
        }
    }

    // Two 16-pixel tiles per wave (pixels wavePixBase+0..15 and +16..31)
    int pA = wavePixBase + n;       pA = (pA < NPIX) ? pA : (NPIX - 1);
    int pB = wavePixBase + 16 + n;  pB = (pB < NPIX) ? pB : (NPIX - 1);
    int woA = pA % WO, tA = pA / WO, hoA = tA % HO, bAi = tA / HO;
    int woB = pB % WO, tB = pB / WO, hoB = tB % HO, bBi = tB / HO;

    v8f accA0 = {}, accA1 = {}, accB0 = {}, accB1 = {};

#pragma unroll
    for (int t = 0; t < 7; ++t) {
        v2f bAv, bBv;
#pragma unroll
        for (int j = 0; j < 2; ++j) {
            int k  = 4 * t + 2 * hiHalf + j;
            int kc = (k < KRED) ? k : 0;
            int c  = kc / KK;
            int r  = kc - c * KK;
            int kh = r / 3;
            int kw = r - kh * 3;
            float va = x[((bAi * CIN + c) * HIN + hoA + kh) * WIN + woA + kw];
            float vb = x[((bBi * CIN + c) * HIN + hoB + kh) * WIN + woB + kw];
            bAv[j] = (k < KRED) ? va : 0.0f;
            bBv[j] = (k < KRED) ? vb : 0.0f;
        }
        v2f a0v = { a0[t][0], a0[t][1] };
        v2f a1v = { a1[t][0], a1[t][1] };
        accA0 = __builtin_amdgcn_wmma_f32_16x16x4_f32(false, a0v, false, bAv, (short)0, accA0, false, false);
        accA1 = __builtin_amdgcn_wmma_f32_16x16x4_f32(false, a1v, false, bAv, (short)0, accA1, false, false);
        accB0 = __builtin_amdgcn_wmma_f32_16x16x4_f32(false, a0v, false, bBv, (short)0, accB0, false, false);
        accB1 = __builtin_amdgcn_wmma_f32_16x16x4_f32(false, a1v, false, bBv, (short)0, accB1, false, false);
    }

    // ---- Scatter offsets (+b1) to LDS ---------------------------------
    {
        const int chBase = hiHalf ? 8 : 0;     // C/D: lane half selects M=j vs M=8+j
        float* offA = &s_off[(waveId * 32 + n) * CO1];
        float* offB = &s_off[(waveId * 32 + 16 + n) * CO1];
#pragma unroll
        for (int j = 0; j < 8; ++j) {
            offA[chBase + j] = accA0[j] + s_b1[chBase + j];
            offB[chBase + j] = accB0[j] + s_b1[chBase + j];
        }
        if (!hiHalf) {                          // channels 16,17 from the 2-row tile
            offA[16] = accA1[0] + s_b1[16];
            offA[17] = accA1[1] + s_b1[17];
            offB[16] = accB1[0] + s_b1[16];
            offB[17] = accB1[1] + s_b1[17];
        }
    }
    __syncthreads();

    // ---- Stage 2: one pixel per lane, bilinear deformable sampling ----
    const int gp = blockPixBase + tid;
    const int p  = (gp < NPIX) ? gp : (NPIX - 1);
    int wo = p % WO, tp = p / WO, ho = tp % HO, b = tp / HO;

    const float* offp = &s_off[tid * CO1];
    const float* xb   = x + (size_t)b * (CIN * HIN * WIN);

    float acc0 = s_b2[0], acc1 = s_b2[1], acc2 = s_b2[2];
#pragma unroll
    for (int kk = 0; kk < KK; ++kk) {
        const int kh = kk / 3, kw = kk - 3 * (kk / 3);
        float dy = offp[2 * kk];
        float dx = offp[2 * kk + 1];
        float py = (float)(ho + kh) + dy;
        float px = (float)(wo + kw) + dx;
        float fy0 = floorf(py), fx0 = floorf(px);
        float wy = py - fy0, wx = px - fx0;

        float v0 = 0.f, v1 = 0.f, v2 = 0.f;
#pragma unroll
        for (int cor = 0; cor < 4; ++cor) {
            float fy  = fy0 + (float)(cor >> 1);
            float fx  = fx0 + (float)(cor & 1);
            float wgt = ((cor >> 1) ? wy : 1.0f - wy) * ((cor & 1) ? wx : 1.0f - wx);
            bool valid = (fy >= 0.0f) && (fy <= (float)(HIN - 1)) &&
                         (fx >= 0.0f) && (fx <= (float)(WIN - 1));
            wgt = valid ? wgt : 0.0f;
            int yi = (int)fminf(fmaxf(fy, 0.0f), (float)(HIN - 1));
            int xi = (int)fminf(fmaxf(fx, 0.0f), (float)(WIN - 1));
            int base = yi * WIN + xi;
            v0 += xb[base]                 * wgt;
            v1 += xb[HIN * WIN + base]     * wgt;
            v2 += xb[2 * HIN * WIN + base] * wgt;
        }
        acc0 += s_w2[     kk] * v0 + s_w2[ 9 + kk] * v1 + s_w2[18 + kk] * v2;
        acc1 += s_w2[27 + kk] * v0 + s_w2[36 + kk] * v1 + s_w2[45 + kk] * v2;
        acc2 += s_w2[54 + kk] * v0 + s_w2[63 + kk] * v1 + s_w2[72 + kk] * v2;
    }

    if (gp < NPIX) {
        out[((b * 3 + 0) * HO + ho) * WO + wo] = acc0;
        out[((b * 3 + 1) * HO + ho) * WO + wo] = acc1;
        out[((b * 3 + 2) * HO + ho) * WO + wo] = acc2;
    }
}

extern "C" void kernel_launch(void* const* d_in, const int* in_sizes, int n_in,
                              void* d_out, int out_size, void* d_ws, size_t ws_size,
                              hipStream_t stream) {
    (void)in_sizes; (void)n_in; (void)out_size; (void)d_ws; (void)ws_size;
    const float* x  = (const float*)d_in[0];
    const float* w1 = (const float*)d_in[1];
    const float* b1 = (const float*)d_in[2];
    const float* w2 = (const float*)d_in[3];
    const float* b2 = (const float*)d_in[4];
    float* out = (float*)d_out;

    const int nblocks = (NPIX + PIX_PER_BLOCK - 1) / PIX_PER_BLOCK;
    deform_conv_fused<<<nblocks, 256, 0, stream>>>(x, w1, b1, w2, b2, out);
}